// BackwardVariableSplitter_80633716015468
// MI455X (gfx1250) — compile-verified
//
#include <hip/hip_runtime.h>
#include <hip/hip_bf16.h>

#define N_VARS   1024
#define EMB      256
#define HID      128
#define LDS_PAD  132   // 132 mod 64 = 4 -> lanes tx hit banks (4*tx + c) % 64, conflict-free; 132*4 bytes % 16 == 0

typedef __attribute__((ext_vector_type(2))) float v2f;
typedef __attribute__((ext_vector_type(8))) float v8f;

// -------------------------------------------------------------------------
// context[c] = relu( sum_e nse[e] * Wc[e,c] + bc[c] ),  Wc row-major [256,128]
// -------------------------------------------------------------------------
__global__ void context_kernel(const float* __restrict__ nse,
                               const float* __restrict__ Wc,
                               const float* __restrict__ bc,
                               float* __restrict__ ctx) {
    int c = threadIdx.x;            // 0..127, coalesced over c
    float acc = bc[c];
    #pragma unroll 8
    for (int e = 0; e < EMB; ++e)
        acc = fmaf(nse[e], Wc[e * HID + c], acc);
    ctx[c] = fmaxf(acc, 0.0f);
}

// -------------------------------------------------------------------------
// C[M,128] = A[M,K] @ B[K,128] + bias0 + bias1   (fp32, row-major, WMMA)
// One wave (32 lanes) per block computes one 16x16 tile via
// V_WMMA_F32_16X16X4_F32, stepping K by 4.
//
// fp32 lane layouts (ISA 7.12.2):
//   A 16x4 : lanes 0-15 -> M=lane, vgpr0=K0, vgpr1=K1 ; lanes 16-31 -> K2,K3
//   B 4x16 : vgpr0 = {K0 row | K2 row}, vgpr1 = {K1 | K3}, N = lane&15
//   C 16x16: vgpr r -> M = r + 8*(lane>=16), N = lane&15
// -------------------------------------------------------------------------
__global__ __launch_bounds__(32)
void wmma_gemm_kernel(const float* __restrict__ A,
                      const float* __restrict__ B,
                      const float* __restrict__ bias0,   // may be null
                      const float* __restrict__ bias1,   // may be null
                      float* __restrict__ C,
                      int K) {
    const int m0   = blockIdx.x * 16;
    const int n0   = blockIdx.y * 16;
    const int lane = threadIdx.x;          // 0..31, all lanes active (EXEC=~0)
    const int half = lane >> 4;            // 0: K0/K1, 1: K2/K3
    const int l    = lane & 15;

    const float* Arow = A + (size_t)(m0 + l) * K;

    v8f acc = {0.f, 0.f, 0.f, 0.f, 0.f, 0.f, 0.f, 0.f};

    for (int k = 0; k < K; k += 4) {
        const int ka = k + half * 2;
        v2f a, b;
        a.x = Arow[ka];
        a.y = Arow[ka + 1];
        b.x = B[(size_t)ka       * HID + n0 + l];
        b.y = B[(size_t)(ka + 1) * HID + n0 + l];
        acc = __builtin_amdgcn_wmma_f32_16x16x4_f32(
                  /*neg_a=*/false, a, /*neg_b=*/false, b,
                  /*c_mod=*/(short)0, acc, /*reuse_a=*/false, /*reuse_b=*/false);
    }

    const int col = n0 + l;
    float bias = 0.0f;
    if (bias0) bias += bias0[col];
    if (bias1) bias += bias1[col];

    #pragma unroll
    for (int r = 0; r < 8; ++r) {
        const int row = m0 + r + half * 8;
        C[(size_t)row * HID + col] = acc[r] + bias;
    }
}

// -------------------------------------------------------------------------
// scores[i,j] = sum_c relu(qp[i,c] + kp[j,c]) * W2[c] + b2      (i < j)
// (b1 was pre-folded into qp by the q_proj GEMM launch.)
// Block = 256 threads = 16x16 (i,j) tile; blocks strictly below the
// diagonal are skipped. Output packed row-major upper-triangular:
//   idx = i*(2n - i - 1)/2 + (j - i - 1)
// -------------------------------------------------------------------------
__global__ __launch_bounds__(256)
void pairwise_kernel(const float* __restrict__ qp,
                     const float* __restrict__ kp,
                     const float* __restrict__ W2,
                     const float* __restrict__ b2,
                     float* __restrict__ out) {
    const int bj = blockIdx.x;   // j tile
    const int bi = blockIdx.y;   // i tile
    if (bi > bj) return;         // entire tile has i > j

    __shared__ float sq[16 * LDS_PAD];
    __shared__ float sk[16 * LDS_PAD];
    __shared__ float sw[HID];

    const int tid = threadIdx.x;

    // Stage 16 q-rows and 16 k-rows (float4, coalesced), plus W2.
    for (int idx = tid; idx < 16 * (HID / 4); idx += 256) {
        const int r  = idx >> 5;          // row within tile
        const int c4 = (idx & 31) << 2;   // column (multiple of 4)
        const float4 vq = *(const float4*)(qp + (size_t)(bi * 16 + r) * HID + c4);
        const float4 vk = *(const float4*)(kp + (size_t)(bj * 16 + r) * HID + c4);
        *(float4*)(sq + r * LDS_PAD + c4) = vq;
        *(float4*)(sk + r * LDS_PAD + c4) = vk;
    }
    if (tid < HID) sw[tid] = W2[tid];
    __syncthreads();

    const int tx = tid & 15;         // j within tile
    const int ty = tid >> 4;         // i within tile
    const int i = bi * 16 + ty;
    const int j = bj * 16 + tx;
    if (i >= j) return;

    const float* q = sq + ty * LDS_PAD;
    const float* k = sk + tx * LDS_PAD;

    float acc = b2[0];
    #pragma unroll
    for (int c = 0; c < HID; c += 4) {
        const float4 a = *(const float4*)(q + c);
        const float4 b = *(const float4*)(k + c);
        const float4 w = *(const float4*)(sw + c);
        acc = fmaf(fmaxf(a.x + b.x, 0.0f), w.x, acc);
        acc = fmaf(fmaxf(a.y + b.y, 0.0f), w.y, acc);
        acc = fmaf(fmaxf(a.z + b.z, 0.0f), w.z, acc);
        acc = fmaf(fmaxf(a.w + b.w, 0.0f), w.w, acc);
    }

    const int base = i * (2 * N_VARS - i - 1) / 2;   // < 2^20, fits int
    out[base + (j - i - 1)] = acc;
}

// -------------------------------------------------------------------------
// Launch
// -------------------------------------------------------------------------
extern "C" void kernel_launch(void* const* d_in, const int* in_sizes, int n_in,
                              void* d_out, int out_size, void* d_ws, size_t ws_size,
                              hipStream_t stream) {
    (void)in_sizes; (void)n_in; (void)out_size; (void)ws_size;

    const float* nse  = (const float*)d_in[0];   // [256]
    const float* prev = (const float*)d_in[1];   // [1024,256]
    const float* Wq   = (const float*)d_in[2];   // [256,128]
    const float* bq   = (const float*)d_in[3];   // [128]
    const float* Wk   = (const float*)d_in[4];   // [256,128]
    const float* bk   = (const float*)d_in[5];   // [128]
    const float* Wc   = (const float*)d_in[6];   // [256,128]
    const float* bc   = (const float*)d_in[7];   // [128]
    const float* W1   = (const float*)d_in[8];   // [256,128] (top 128 rows = q side, bottom = k side)
    const float* b1   = (const float*)d_in[9];   // [128]
    const float* W2   = (const float*)d_in[10];  // [128,1]
    const float* b2   = (const float*)d_in[11];  // [1]
    float* out = (float*)d_out;                  // [523776]

    // Workspace carve-up (floats): ctx | queries | keys | q_proj | k_proj
    float* ws      = (float*)d_ws;
    float* ctx     = ws;                          // 128
    float* queries = ctx + HID;                   // 1024*128
    float* keys    = queries + N_VARS * HID;      // 1024*128
    float* q_proj  = keys    + N_VARS * HID;      // 1024*128
    float* k_proj  = q_proj  + N_VARS * HID;      // 1024*128
    // total: 128 + 4*131072 floats  ~= 2.1 MB

    // 1) context = relu(nse @ Wc + bc)
    context_kernel<<<1, HID, 0, stream>>>(nse, Wc, bc, ctx);

    // 2) queries = prev @ Wq + bq + ctx ; keys = prev @ Wk + bk + ctx
    dim3 ggrid(N_VARS / 16, HID / 16);
    wmma_gemm_kernel<<<ggrid, 32, 0, stream>>>(prev, Wq, bq, ctx, queries, EMB);
    wmma_gemm_kernel<<<ggrid, 32, 0, stream>>>(prev, Wk, bk, ctx, keys,    EMB);

    // 3) q_proj = queries @ W1[:128] + b1  (b1 folded here)
    //    k_proj = keys    @ W1[128:]
    wmma_gemm_kernel<<<ggrid, 32, 0, stream>>>(queries, W1,             b1,      nullptr, q_proj, HID);
    wmma_gemm_kernel<<<ggrid, 32, 0, stream>>>(keys,    W1 + HID * HID, nullptr, nullptr, k_proj, HID);

    // 4) pairwise upper-triangular scores
    dim3 pgrid(N_VARS / 16, N_VARS / 16);
    pairwise_kernel<<<pgrid, 256, 0, stream>>>(q_proj, k_proj, W2, b2, out);
}